// LowShelfFilter_56143812493978
// MI455X (gfx1250) — compile-verified
//
#include <hip/hip_runtime.h>
#include <cmath>
#include <cstdint>

#ifndef M_PI
#define M_PI 3.14159265358979323846
#endif

// ---- Problem constants (fixed by the reference) ----
constexpr int B = 32;
constexpr int T = 960000;

// ---- Parallel-IIR decomposition ----
// Stable biquad: state influence decays as poleRadius^n, so each chunk can
// recover its initial state by a short zero-seeded warm-up over the W samples
// preceding it (chunk 0 is exact). One pass, no inter-chunk scan.
constexpr int L = 960;               // samples per chunk; T = C*L exactly
constexpr int C = T / L;             // 1000 chunks per batch row
constexpr int TILE = 32;             // samples staged per LDS tile
constexpr int SEG_STRIDE = TILE + 4; // 36 floats = 144 B: 16B-aligned, bank-skewed
constexpr int WAVES = 4;             // waves per workgroup (wave32)
constexpr int BLOCK = WAVES * 32;    // 128 threads
constexpr int NCHUNK = B * C;        // 32000 chunk-threads (250 blocks exactly)
constexpr int UPL = TILE / 4;        // 8 B128 units per lane per tile

typedef __attribute__((address_space(3))) uint8_t lds_u8_t;
typedef float v4f __attribute__((ext_vector_type(4)));

__global__ __launch_bounds__(BLOCK) void biquad_chunks(
    const float* __restrict__ x, float* __restrict__ y,
    float b0, float b1, float b2, float a1, float a2,
    int W, int WT, int NT)
{
  // [buf][wave][segment(=lane's chunk)][sample] ; 2*4*32*36*4 = 36864 B
  __shared__ __align__(16) float lds[2][WAVES][32][SEG_STRIDE];

  const int lane = threadIdx.x & 31;
  const int wv   = threadIdx.x >> 5;
  const int g    = blockIdx.x * BLOCK + threadIdx.x; // global chunk id < B*C
  const int b    = g / C;
  const int c    = g - b * C;

  // Byte offsets into x (all < 2^27, fit in int32).
  const int startEl   = b * T + c * L;          // first sample of this chunk
  const int rowStartB = b * (T * 4);            // row begin (clamp floor)
  const int regionB   = startEl * 4 - W * 4;    // staged region: [start-W, start+L)

  // Raw 32-bit LDS byte address of the shared buffer (addrspace(3) ptrtoint),
  // needed as the async-DMA destination operand.
  const uint32_t ldsBase   = (uint32_t)(uintptr_t)(lds_u8_t*)(&lds[0][0][0][0]);
  const uint32_t bufStride = (uint32_t)(WAVES * 32 * SEG_STRIDE * 4);
  const uint32_t waveOff   = (uint32_t)wv * (uint32_t)(32 * SEG_STRIDE * 4);
  const int u16 = (lane & 7) * 16;  // this lane's 16B unit within a segment
  const int sHi = lane >> 3;        // which of 4 segments this lane serves per step

  // Hoisted (tile-invariant) cooperative-load descriptors: for each of the 8
  // units this lane stages per tile, the owning segment's global base (+unit
  // offset), its clamp floor, and its LDS destination address.
  int segB[UPL];        // global byte base of unit (region base + u16)
  int rowM[UPL];        // clamp floor (row start) for that segment
  uint32_t ldsU[UPL];   // LDS byte address of unit in buffer 0
#pragma unroll
  for (int j = 0; j < UPL; ++j) {
    const int s = 4 * j + sHi;                      // segment index 0..31
    segB[j] = __shfl(regionB, s) + u16;
    rowM[j] = __shfl(rowStartB, s);
    ldsU[j] = ldsBase + waveOff + (uint32_t)(s * (SEG_STRIDE * 4)) + (uint32_t)u16;
  }

  // Cooperatively stage one TILE for all 32 chunks owned by this wave:
  // 32 segments x 8 B128 units; each instruction covers 4 runs of 128
  // contiguous bytes. Per-wave (no workgroup barrier needed: ASYNCcnt is
  // wave state and covers all 32 lanes' DMAs).
  auto issue_tile = [&](int tt) {
    // Reused buffer was fully consumed last iteration; make sure its DS reads
    // have drained before the DMA engine may overwrite it.
    asm volatile("s_wait_dscnt 0x0" ::: "memory");
    const uint32_t bufOff = (uint32_t)(tt & 1) * bufStride;
    const int tb = tt * (TILE * 4);
#pragma unroll
    for (int j = 0; j < UPL; ++j) {
      int goff = segB[j] + tb;
      if (goff < rowM[j]) goff = rowM[j];           // clamp (c==0 warm-up)
      const uint32_t la = ldsU[j] + bufOff;
      // CDNA5 async DMA: global (saddr + 32b voffset) -> LDS, ASYNCcnt-tracked.
      asm volatile("global_load_async_to_lds_b128 %0, %1, %2"
                   :: "v"(la), "v"(goff), "s"(x)
                   : "memory");
    }
  };

  float s1 = 0.f, s2 = 0.f;
  float* __restrict__ yp = y + startEl;

  issue_tile(0);
  for (int tt = 0; tt < NT; ++tt) {
    if (tt + 1 < NT) {
      issue_tile(tt + 1);                               // double-buffered prefetch
      asm volatile("s_wait_asynccnt 0x8" ::: "memory"); // tile tt complete
    } else {
      asm volatile("s_wait_asynccnt 0x0" ::: "memory");
    }
    const float* seg = &lds[tt & 1][wv][lane][0];

    // Chunk 0 starts from the true zero state: discard warm-up garbage.
    if (c == 0 && tt == WT) { s1 = 0.f; s2 = 0.f; }

    if (tt >= WT) {
      float* out = yp + (tt - WT) * TILE;               // 16B-aligned
#pragma unroll
      for (int i4 = 0; i4 < TILE; i4 += 4) {
        v4f o;
#pragma unroll
        for (int k = 0; k < 4; ++k) {
          const float xv = seg[i4 + k];
          const float yv = fmaf(b0, xv, s1);            // y  = b0*x + s1
          s1 = fmaf(b1, xv, fmaf(-a1, yv, s2));         // s1 = b1*x - a1*y + s2
          s2 = fmaf(b2, xv, -a2 * yv);                  // s2 = b2*x - a2*y
          o[k] = yv;
        }
        // y is write-once: non-temporal B128 store keeps x resident in L2
        // (warm-up re-reads want x cached; y should stream through).
        __builtin_nontemporal_store(o, (v4f*)(out + i4));
      }
    } else {
#pragma unroll
      for (int i = 0; i < TILE; ++i) {                  // warm-up: no stores
        const float xv = seg[i];
        const float yv = fmaf(b0, xv, s1);
        s1 = fmaf(b1, xv, fmaf(-a1, yv, s2));
        s2 = fmaf(b2, xv, -a2 * yv);
      }
    }
  }
}

__global__ void copy_lengths(const int* __restrict__ src, int* __restrict__ dst) {
  if (threadIdx.x < B) dst[threadIdx.x] = src[threadIdx.x];
}

extern "C" void kernel_launch(void* const* d_in, const int* in_sizes, int n_in,
                              void* d_out, int out_size, void* d_ws, size_t ws_size,
                              hipStream_t stream) {
  (void)in_sizes; (void)n_in; (void)out_size; (void)d_ws; (void)ws_size;

  // W3 cookbook low-shelf coefficients, double on host, cast to f32 like the ref.
  const double w0 = 2.0 * M_PI * 1000.0 / 16000.0;
  const double A  = pow(10.0, 6.0 / 40.0);
  const double al = sin(w0) / (2.0 * 0.7);
  const double cw = cos(w0);
  const double sA = sqrt(A);
  const double a0 = (A + 1.0) + (A - 1.0) * cw + 2.0 * sA * al;
  const float b0 = (float)((A * ((A + 1.0) - (A - 1.0) * cw + 2.0 * sA * al)) / a0);
  const float b1 = (float)((2.0 * A * ((A - 1.0) - (A + 1.0) * cw)) / a0);
  const float b2 = (float)((A * ((A + 1.0) - (A - 1.0) * cw - 2.0 * sA * al)) / a0);
  const float a1 = (float)((-2.0 * ((A - 1.0) + (A + 1.0) * cw)) / a0);
  const float a2 = (float)(((A + 1.0) + (A - 1.0) * cw - 2.0 * sA * al) / a0);

  // Warm-up length from the true pole radius: r^W < 1e-12.
  const double da1 = (double)a1, da2 = (double)a2;
  const double disc = da1 * da1 - 4.0 * da2;
  double r;
  if (disc < 0.0) r = sqrt(fmax(da2, 1e-30));
  else { const double sd = sqrt(disc);
         r = fmax(fabs((-da1 + sd) * 0.5), fabs((-da1 - sd) * 0.5)); }
  if (r < 1e-3) r = 1e-3;
  if (r > 0.999999) r = 0.999999;
  int W = (int)ceil(log(1e-12) / log(r));
  W = ((W + TILE - 1) / TILE) * TILE;  // whole tiles
  if (W < TILE) W = TILE;
  if (W > L) W = L;
  const int WT = W / TILE;
  const int NT = WT + L / TILE;

  const float* x   = (const float*)d_in[0];
  const int* lens  = (const int*)d_in[1];
  float* y         = (float*)d_out;

  biquad_chunks<<<NCHUNK / BLOCK, BLOCK, 0, stream>>>(x, y, b0, b1, b2, a1, a2,
                                                      W, WT, NT);
  // Second reference output: pass input_lengths through (bit copy).
  copy_lengths<<<1, 32, 0, stream>>>(lens, (int*)d_out + (size_t)B * T);
}